// FRAP_47966194761910
// MI455X (gfx1250) — compile-verified
//
#include <hip/hip_runtime.h>
#include <hip/hip_bf16.h>

typedef __attribute__((ext_vector_type(2))) float v2f;
typedef __attribute__((ext_vector_type(8))) float v8f;

#define NTHREADS 256

__device__ __forceinline__ float lrelu(float v) { return v > 0.0f ? v : 0.01f * v; }

// MASK_DATA (8 rows x 7 cols) flattened; reference reshapes to [7][8], so the
// mask value at pixel p (p = h*8+w, 0..55) is simply c_mask[p].
__constant__ float c_mask[56] = {
    0.5f, 0.5f, 1.0f, 1.0f, 1.0f, 1.0f, 1.0f,
    0.5f, 1.0f, 0.5f, 1.0f, 1.0f, 1.0f, 1.0f,
    0.5f, 1.0f, 0.5f, 1.0f, 1.0f, 1.0f, 1.0f,
    1.0f, 0.5f, 0.5f, 1.0f, 1.0f, 1.0f, 1.0f,
    1.0f, 1.0f, 1.0f, 1.0f, 0.5f, 0.5f, 1.0f,
    1.0f, 1.0f, 1.0f, 1.0f, 0.5f, 1.0f, 0.5f,
    1.0f, 1.0f, 1.0f, 1.0f, 0.5f, 1.0f, 0.5f,
    1.0f, 1.0f, 1.0f, 1.0f, 1.0f, 0.5f, 0.5f
};
__constant__ int c_pairA[8] = {0, 0, 4, 1, 2, 2, 6, 3};
__constant__ int c_pairB[8] = {4, 1, 5, 5, 6, 3, 7, 7};

// Stage weight [M,K] (row-major) + bias[M] into zero-padded LDS tiles.
// Executed by waves 1..7 only (224 threads), concurrently with the recurrence.
__device__ __forceinline__ void stage224(const float* __restrict__ Wg,
                                         const float* __restrict__ bg,
                                         int M, int K,
                                         float (*Wl)[32], float* Bias, int stid) {
  for (int i = stid; i < 32 * 32; i += 224) {
    int r = i >> 5, c = i & 31;
    Wl[r][c] = (r < M && c < K) ? Wg[r * K + c] : 0.0f;
  }
  if (stid < 32) Bias[stid] = (stid < M) ? bg[stid] : 0.0f;  // padded rows -> 0
}

// Out[32][64] = lrelu(Wl[32x32] @ In[32][64] + Bias) [* Scale], via
// v_wmma_f32_16x16x4_f32. 8 waves: wave w -> 16x16 tile (row w>>2, col w&3).
__device__ __forceinline__ void mm16x16x4(const float (*In)[64], float (*Out)[64],
                                          const float (*Wl)[32], const float* Bias,
                                          int Kp, const float (*Scale)[64], int tid) {
  const int lane = tid & 31;
  const int wave = tid >> 5;
  const int mt   = wave >> 2;           // 0..1
  const int nt   = wave & 3;            // 0..3
  const int ln16 = lane & 15;
  const int kg   = (lane >> 4) << 1;    // 0 or 2: K sub-offset per ISA A/B layout

  v8f acc = {0.0f, 0.0f, 0.0f, 0.0f, 0.0f, 0.0f, 0.0f, 0.0f};
  const int arow = mt * 16 + ln16;      // A: lane = M + 16*(K>=2), vgpr = K&1
  const int bcol = nt * 16 + ln16;      // B: lane = N + 16*(K>=2), vgpr = K&1

  for (int k0 = 0; k0 < Kp; k0 += 4) {
    v2f a, b;
    a.x = Wl[arow][k0 + kg];
    a.y = Wl[arow][k0 + kg + 1];
    b.x = In[k0 + kg][bcol];
    b.y = In[k0 + kg + 1][bcol];
    acc = __builtin_amdgcn_wmma_f32_16x16x4_f32(
        /*neg_a=*/false, a, /*neg_b=*/false, b,
        /*c_mod=*/(short)0, acc, /*reuse_a=*/false, /*reuse_b=*/false);
  }

  // C/D layout: col = lane&15, rows = 8*(lane>=16) + vgpr
  const int rowbase = mt * 16 + ((lane >> 4) << 3);
#pragma unroll
  for (int v = 0; v < 8; ++v) {
    float val = lrelu(acc[v] + Bias[rowbase + v]);
    if (Scale) val *= Scale[rowbase + v][bcol];   // fused r = h*m epilogue
    Out[rowbase + v][bcol] = val;
  }
}

__device__ __forceinline__ void conv(const float (*In)[64], float (*Out)[64],
                                     const float (*Wl)[32], const float* Bias,
                                     int K, const float (*Scale)[64], int tid) {
  mm16x16x4(In, Out, Wl, Bias, (K + 3) & ~3, Scale, tid);
  __syncthreads();
}

__global__ __launch_bounds__(NTHREADS)
void frap_fused(const float* __restrict__ x,
                const float* __restrict__ Wv1, const float* __restrict__ bv1,
                const float* __restrict__ Wv2, const float* __restrict__ bv2,
                const float* __restrict__ Wp1, const float* __restrict__ bp1,
                const float* __restrict__ Wp2, const float* __restrict__ bp2,
                const float* __restrict__ We,  const float* __restrict__ be,
                const float* __restrict__ Cp1, const float* __restrict__ cbp1,
                const float* __restrict__ Cp2, const float* __restrict__ cbp2,
                const float* __restrict__ Cm1, const float* __restrict__ cbm1,
                const float* __restrict__ Cm2, const float* __restrict__ cbm2,
                const float* __restrict__ Cm3, const float* __restrict__ cbm3,
                const float* __restrict__ Cc1, const float* __restrict__ cbc1,
                const float* __restrict__ Cc2, const float* __restrict__ cbc2,
                float* __restrict__ out) {
  __shared__ float B0[32][64], B1[32][64], B2[32][64];
  __shared__ float Wm1[32][32], Wm2[32][32], Wm3[32][32];
  __shared__ float Wq1[32][32], Wq2[32][32], Wc1l[32][32], Wc2l[32][32];
  __shared__ float Bm1[32], Bm2[32], Bm3[32], Bq1[32], Bq2[32], Bc1l[32], Bc2l[32];
  __shared__ float dem[8][16];

  const int tid = threadIdx.x;

  // ---- Phase A (overlapped): waves 1..7 stage all weights + mask input,
  //      wave 0 runs the serial demand recurrence via register shuffles. ----
  if (tid >= 32) {
    const int stid = tid - 32;  // 0..223
    stage224(Cm1, cbm1, 4, 1,  Wm1,  Bm1,  stid);
    stage224(Cm2, cbm2, 20, 4, Wm2,  Bm2,  stid);
    stage224(Cm3, cbm3, 20, 20, Wm3, Bm3,  stid);
    stage224(Cp1, cbp1, 20, 32, Wq1, Bq1,  stid);
    stage224(Cp2, cbp2, 20, 20, Wq2, Bq2,  stid);
    stage224(Cc1, cbc1, 8, 20, Wc1l, Bc1l, stid);
    stage224(Cc2, cbc2, 1, 8,  Wc2l, Bc2l, stid);
    // mask-branch input: row 0 = MASK pixels, rows 1..31 / cols 56..63 zero
    for (int i = stid; i < 32 * 64; i += 224) {
      int rr = i >> 6, cc = i & 63;
      B1[rr][cc] = (rr == 0 && cc < 56) ? c_mask[cc] : 0.0f;
    }
  } else {
    // wave 0: 8-step recurrence, loop-carried state in registers (lanes 0..15)
    float e = (tid < 16) ? x[tid] : 0.0f;
    for (int i = 0; i < 8; ++i) {
      float sv = __shfl(e, i, 32);       // cur[i]     (reference's cur=emb bug)
      float sp = __shfl(e, 8 + i, 32);   // cur[8+i]
      if (tid < 16) {
        float hv0 = lrelu(sv * Wv1[0] + bv1[0]);
        float hv1 = lrelu(sv * Wv1[1] + bv1[1]);
        float hp0 = lrelu(sp * Wp1[0] + bp1[0]);
        float hp1 = lrelu(sp * Wp1[1] + bp1[1]);
        float f[8];
#pragma unroll
        for (int o = 0; o < 4; ++o) {
          f[o]     = lrelu(hv0 * Wv2[2 * o] + hv1 * Wv2[2 * o + 1] + bv2[o]);
          f[4 + o] = lrelu(hp0 * Wp2[2 * o] + hp1 * Wp2[2 * o + 1] + bp2[o]);
        }
        float acc = be[tid];
#pragma unroll
        for (int c = 0; c < 8; ++c) acc += We[tid * 8 + c] * f[c];
        e = lrelu(acc);
        dem[i][tid] = e;                 // write-only here; read after barrier
      }
    }
  }
  __syncthreads();

  // ---- Phase B: build conv input. B0[ch][p] = t_flat[ch*56+p], where
  //      t[r][j][c] = pd[r + (r>=j)][c] (c<16) | pd[j][c-16], and
  //      pd[s][cc] = dem[pairA[s]][cc] + dem[pairB[s]][cc] computed on the fly.
  for (int idx = tid; idx < 32 * 64; idx += NTHREADS) {
    int ch = idx >> 6, col = idx & 63;
    float v = 0.0f;
    if (col < 56) {
      int fidx = ch * 56 + col;           // flat index into [7][8][32]
      int r = fidx >> 8;                  // /256
      int j = (fidx >> 5) & 7;            // /32 % 8
      int c = fidx & 31;
      int src = (c < 16) ? (r + ((r >= j) ? 1 : 0)) : j;
      int cc  = c & 15;
      v = dem[c_pairA[src]][cc] + dem[c_pairB[src]][cc];
    }
    B0[ch][col] = v;
  }
  __syncthreads();

  // ---- Phase C: seven WMMA conv layers (uniform across all 8 waves) ----
  conv(B1, B2, Wm1, Bm1, 1,  nullptr, tid);   // mask 1->4
  conv(B2, B1, Wm2, Bm2, 4,  nullptr, tid);   // mask 4->20
  conv(B1, B2, Wm3, Bm3, 20, nullptr, tid);   // mask 20->20  (m -> B2)
  conv(B0, B1, Wq1, Bq1, 32, nullptr, tid);   // main 32->20
  conv(B1, B0, Wq2, Bq2, 20, B2, tid);        // main 20->20, fused *m -> B0
  conv(B0, B1, Wc1l, Bc1l, 20, nullptr, tid); // comp 20->8
  conv(B1, B2, Wc2l, Bc2l, 8,  nullptr, tid); // comp 8->1    ([1,56] in B2 row 0)

  // ---- Phase D: sum over H (7 rows of 8) -> out[8] ----
  if (tid < 8) {
    float s = 0.0f;
#pragma unroll
    for (int h = 0; h < 7; ++h) s += B2[0][h * 8 + tid];
    out[tid] = s;
  }
}

extern "C" void kernel_launch(void* const* d_in, const int* in_sizes, int n_in,
                              void* d_out, int out_size, void* d_ws, size_t ws_size,
                              hipStream_t stream) {
  (void)in_sizes; (void)n_in; (void)out_size; (void)d_ws; (void)ws_size;
  const float* p[25];
  for (int i = 0; i < 25; ++i) p[i] = (const float*)d_in[i];
  frap_fused<<<dim3(1), dim3(NTHREADS), 0, stream>>>(
      p[0], p[1], p[2], p[3], p[4], p[5], p[6], p[7], p[8], p[9], p[10],
      p[11], p[12], p[13], p[14], p[15], p[16], p[17], p[18], p[19], p[20],
      p[21], p[22], p[23], p[24], (float*)d_out);
}